// DifferentialMultiHeadAttentionWithHierarchicalMask_51711406244188
// MI455X (gfx1250) — compile-verified
//
#include <hip/hip_runtime.h>
#include <cstddef>

// Problem constants (match reference)
static constexpr int B  = 4;
static constexpr int S  = 1024;
static constexpr int D  = 1024;
static constexpr int H  = 16;
static constexpr int DH = 64;
static constexpr float GN_EPS = 1e-5f;
static constexpr float OUT_SCALE = 0.19999999999999996f; // 1 - lambda_init(0.8)

typedef __attribute__((ext_vector_type(2))) float v2f;
typedef __attribute__((ext_vector_type(8))) float v8f;

// D = A(16x4) * B(4x16) + C(16x16), all fp32, wave32.
// 8-arg form: (neg_a, A, neg_b, B, c_mod, C, reuse_a, reuse_b)
#define WMMA_F32X4(a, b, c) \
  __builtin_amdgcn_wmma_f32_16x16x4_f32(false, (a), false, (b), (short)0, (c), false, false)

// ---------------------------------------------------------------------------
// Generic GEMM: Y[M,N] = X[M,K] * W[N,K]^T + bias[N]
// One wave -> 16(M) x 64(N) tile, 4 accumulators sharing the A fragment.
// Double-buffered K-groups of 16 (4 WMMA steps): all 20 loads of group G+1
// are issued before the 16 WMMAs of group G, so s_wait_loadcnt is partial
// and L2 latency hides behind matrix math.  blockDim.x = 128 (4 waves).
// ---------------------------------------------------------------------------
__global__ __launch_bounds__(128) void gemm_bias_wmma(
    const float* __restrict__ X, const float* __restrict__ W,
    const float* __restrict__ bias, float* __restrict__ Y,
    int M, int N, int K) {
  const int wave = blockIdx.x * 4 + (threadIdx.x >> 5);
  const int lane = threadIdx.x & 31;
  const int tiles_n = N >> 6;
  const int total = (M >> 4) * tiles_n;
  if (wave >= total) return;
  const int tm = wave / tiles_n;
  const int tn = wave - tm * tiles_n;
  const int m0 = tm << 4, n0 = tn << 6;
  const int nlo = lane & 15;
  const int koff = (lane >> 4) << 1;  // lanes 16-31 hold K+2,K+3

  const float* xrow = X + (size_t)(m0 + nlo) * K + koff;
  const float* w0 = W + (size_t)(n0 + nlo +  0) * K + koff;
  const float* w1 = W + (size_t)(n0 + nlo + 16) * K + koff;
  const float* w2 = W + (size_t)(n0 + nlo + 32) * K + koff;
  const float* w3 = W + (size_t)(n0 + nlo + 48) * K + koff;

  v8f acc0 = {}, acc1 = {}, acc2 = {}, acc3 = {};

  // Fragment group: 4 K-steps (16 K-elements). Two buffers, 2-deep pipeline.
  v2f aA[4], bA[4][4], aB[4], bB[4][4];

  auto load_group = [&](v2f (&af)[4], v2f (&bf)[4][4], int kk) {
#pragma unroll
    for (int u = 0; u < 4; ++u) {
      const int o = kk + 4 * u;
      af[u]    = *(const v2f*)(xrow + o);
      bf[u][0] = *(const v2f*)(w0 + o);
      bf[u][1] = *(const v2f*)(w1 + o);
      bf[u][2] = *(const v2f*)(w2 + o);
      bf[u][3] = *(const v2f*)(w3 + o);
    }
  };
  auto mma_group = [&](v2f (&af)[4], v2f (&bf)[4][4]) {
#pragma unroll
    for (int u = 0; u < 4; ++u) {
      acc0 = WMMA_F32X4(af[u], bf[u][0], acc0);
      acc1 = WMMA_F32X4(af[u], bf[u][1], acc1);
      acc2 = WMMA_F32X4(af[u], bf[u][2], acc2);
      acc3 = WMMA_F32X4(af[u], bf[u][3], acc3);
    }
  };

  // 64 groups (K=1024). Pipeline: preload g0; each loop iter consumes 2
  // groups while preloading the next 2; epilogue drains the last pair.
  load_group(aA, bA, 0);
  int kk = 0;
  for (; kk < K - 32; kk += 32) {
    load_group(aB, bB, kk + 16);
    mma_group(aA, bA);
    load_group(aA, bA, kk + 32);
    mma_group(aB, bB);
  }
  load_group(aB, bB, K - 16);
  mma_group(aA, bA);
  mma_group(aB, bB);

  const int col = n0 + nlo;
  const int rbase = m0 + ((lane >> 4) << 3);  // +0 or +8
  const float bi0 = bias[col], bi1 = bias[col + 16];
  const float bi2 = bias[col + 32], bi3 = bias[col + 48];
#pragma unroll
  for (int r = 0; r < 8; ++r) {
    float* yd = Y + (size_t)(rbase + r) * N + col;
    yd[0]  = acc0[r] + bi0;
    yd[16] = acc1[r] + bi1;
    yd[32] = acc2[r] + bi2;
    yd[48] = acc3[r] + bi3;
  }
}

// ---------------------------------------------------------------------------
// Scores: per-(b,h) 16x16 tile of q1.k1^T/8 and q2.k2^T/8, padding-masked,
// written as RAW scores into the attn1/attn2 output regions.
// DH=64 -> 4 groups of 4 K-steps, straight-line 2-deep pipeline.
// ---------------------------------------------------------------------------
__global__ __launch_bounds__(128) void scores_kernel(
    const float* __restrict__ qp, const float* __restrict__ kp,
    const int* __restrict__ pad,
    float* __restrict__ attn1, float* __restrict__ attn2) {
  const int wave = blockIdx.x * 4 + (threadIdx.x >> 5);
  const int lane = threadIdx.x & 31;
  const int TT = S / 16;  // 64 tiles per dim
  const int kj = wave % TT;
  int t = wave / TT;
  const int qi = t % TT;
  const int bh = t / TT;
  const int b = bh / H, h = bh % H;
  const int q0 = qi << 4, k0 = kj << 4;
  const int nlo = lane & 15;
  const int koff = (lane >> 4) << 1;

  const float* qrow = qp + (size_t)(b * S + q0 + nlo) * (2 * D) + h * DH + koff;
  const float* krow = kp + (size_t)(b * S + k0 + nlo) * (2 * D) + h * DH + koff;

  v8f s1 = {}, s2 = {};

  v2f gA[4][4], gB[4][4];  // [step][a1,c1,a2,c2]

  auto load_group = [&](v2f (&g)[4][4], int kk) {
#pragma unroll
    for (int u = 0; u < 4; ++u) {
      const int o = kk + 4 * u;
      g[u][0] = *(const v2f*)(qrow + o);
      g[u][1] = *(const v2f*)(krow + o);
      g[u][2] = *(const v2f*)(qrow + D + o);
      g[u][3] = *(const v2f*)(krow + D + o);
    }
  };
  auto mma_group = [&](v2f (&g)[4][4]) {
#pragma unroll
    for (int u = 0; u < 4; ++u) {
      s1 = WMMA_F32X4(g[u][0], g[u][1], s1);
      s2 = WMMA_F32X4(g[u][2], g[u][3], s2);
    }
  };

  load_group(gA, 0);
  load_group(gB, 16);
  mma_group(gA);
  load_group(gA, 32);
  mma_group(gB);
  load_group(gB, 48);
  mma_group(gA);
  mma_group(gB);

  const int col = k0 + nlo;
  const bool masked = (pad[b * S + col] == 0);
  const int rbase = q0 + ((lane >> 4) << 3);
  const size_t base = (size_t)bh * S * S;
#pragma unroll
  for (int r = 0; r < 8; ++r) {
    size_t off = base + (size_t)(rbase + r) * S + col;
    attn1[off] = masked ? -1e9f : s1[r] * 0.125f;
    attn2[off] = masked ? -1e9f : s2[r] * 0.125f;
  }
}

// ---------------------------------------------------------------------------
// In-place row softmax over both attention maps. One 256-thread block per row.
// ---------------------------------------------------------------------------
__device__ inline void softmax_row(float* __restrict__ row, float* sdata, int tid) {
  float vals[4];
  float m = -3.4e38f;
#pragma unroll
  for (int i = 0; i < 4; ++i) { vals[i] = row[tid + i * 256]; m = fmaxf(m, vals[i]); }
  sdata[tid] = m; __syncthreads();
  for (int s = 128; s > 0; s >>= 1) {
    if (tid < s) sdata[tid] = fmaxf(sdata[tid], sdata[tid + s]);
    __syncthreads();
  }
  const float rowmax = sdata[0]; __syncthreads();
  float sum = 0.f;
#pragma unroll
  for (int i = 0; i < 4; ++i) { vals[i] = expf(vals[i] - rowmax); sum += vals[i]; }
  sdata[tid] = sum; __syncthreads();
  for (int s = 128; s > 0; s >>= 1) {
    if (tid < s) sdata[tid] += sdata[tid + s];
    __syncthreads();
  }
  const float inv = 1.f / sdata[0]; __syncthreads();
#pragma unroll
  for (int i = 0; i < 4; ++i) row[tid + i * 256] = vals[i] * inv;
}

__global__ __launch_bounds__(256) void softmax_kernel(float* __restrict__ attn1,
                                                      float* __restrict__ attn2) {
  __shared__ float sdata[256];
  const size_t rowid = blockIdx.x;
  softmax_row(attn1 + rowid * S, sdata, threadIdx.x);
  __syncthreads();
  softmax_row(attn2 + rowid * S, sdata, threadIdx.x);
}

// ---------------------------------------------------------------------------
// Context: ctx[b,h,q,:] = (C * (attn1 - lambda*attn2)) @ V.
// One wave -> 16 q-rows x full DH=64 for one head; A frag built on the fly.
// Double-buffered K-groups of 8 (2 WMMA steps, 14 loads each).
// ---------------------------------------------------------------------------
__global__ __launch_bounds__(128) void context_kernel(
    const float* __restrict__ attn1, const float* __restrict__ attn2,
    const float* __restrict__ Cmask, const float* __restrict__ V,
    const float* __restrict__ lam, float* __restrict__ ctx) {
  const int wave = blockIdx.x * 4 + (threadIdx.x >> 5);
  const int lane = threadIdx.x & 31;
  const int TQ = S / 16;
  const int qi = wave % TQ;
  const int bh = wave / TQ;
  const int b = bh / H, h = bh % H;
  const float lambda = lam[0];
  const int q0 = qi << 4;
  const int nlo = lane & 15;
  const int koff = (lane >> 4) << 1;

  const int arow = q0 + nlo;
  const float* a1p = attn1 + ((size_t)bh * S + arow) * S + koff;
  const float* a2p = attn2 + ((size_t)bh * S + arow) * S + koff;
  const float* cmp = Cmask + ((size_t)b * S + arow) * S + koff;
  const float* vb = V + (size_t)b * S * D + h * DH + nlo;

  v8f acc0 = {}, acc1 = {}, acc2 = {}, acc3 = {};

  struct Grp { v2f p1[2], p2[2], pc[2], bf[2][4]; };
  Grp A, Bg;

  auto load_group = [&](Grp& g, int kk) {
#pragma unroll
    for (int u = 0; u < 2; ++u) {
      const int o = kk + 4 * u;
      g.p1[u] = *(const v2f*)(a1p + o);
      g.p2[u] = *(const v2f*)(a2p + o);
      g.pc[u] = *(const v2f*)(cmp + o);
      const int kr = o + koff;
      const float* r0 = vb + (size_t)kr * D;
      const float* r1 = vb + (size_t)(kr + 1) * D;
      g.bf[u][0].x = r0[0];  g.bf[u][0].y = r1[0];
      g.bf[u][1].x = r0[16]; g.bf[u][1].y = r1[16];
      g.bf[u][2].x = r0[32]; g.bf[u][2].y = r1[32];
      g.bf[u][3].x = r0[48]; g.bf[u][3].y = r1[48];
    }
  };
  auto mma_group = [&](Grp& g) {
#pragma unroll
    for (int u = 0; u < 2; ++u) {
      v2f a = g.pc[u] * (g.p1[u] - lambda * g.p2[u]);
      acc0 = WMMA_F32X4(a, g.bf[u][0], acc0);
      acc1 = WMMA_F32X4(a, g.bf[u][1], acc1);
      acc2 = WMMA_F32X4(a, g.bf[u][2], acc2);
      acc3 = WMMA_F32X4(a, g.bf[u][3], acc3);
    }
  };

  // 128 groups of 8 (S=1024). 2-deep pipeline, pairs per iteration.
  load_group(A, 0);
  int kk = 0;
  for (; kk < S - 16; kk += 16) {
    load_group(Bg, kk + 8);
    mma_group(A);
    load_group(A, kk + 16);
    mma_group(Bg);
  }
  load_group(Bg, S - 8);
  mma_group(A);
  mma_group(Bg);

  const int rbase = q0 + ((lane >> 4) << 3);
#pragma unroll
  for (int r = 0; r < 8; ++r) {
    float* cd = ctx + ((size_t)b * S + rbase + r) * D + h * DH + nlo;
    cd[0]  = acc0[r];
    cd[16] = acc1[r];
    cd[32] = acc2[r];
    cd[48] = acc3[r];
  }
}

// ---------------------------------------------------------------------------
// GroupNorm stats per (b,h) over (S, DH); stores mean + rsqrt(var+eps).
// ---------------------------------------------------------------------------
__global__ __launch_bounds__(256) void gn_stats_kernel(const float* __restrict__ ctx,
                                                       float* __restrict__ stats) {
  const int bh = blockIdx.x;
  const int b = bh / H, h = bh % H;
  const int tid = threadIdx.x;
  float sum = 0.f, sumsq = 0.f;
  for (int i = tid; i < S * DH; i += 256) {
    const int s = i >> 6, d = i & (DH - 1);
    const float x = ctx[((size_t)b * S + s) * D + h * DH + d];
    sum += x; sumsq += x * x;
  }
  __shared__ float s1[256], s2[256];
  s1[tid] = sum; s2[tid] = sumsq; __syncthreads();
  for (int s = 128; s > 0; s >>= 1) {
    if (tid < s) { s1[tid] += s1[tid + s]; s2[tid] += s2[tid + s]; }
    __syncthreads();
  }
  if (tid == 0) {
    const float inv_n = 1.0f / (float)(S * DH);
    const float mean = s1[0] * inv_n;
    const float var = s2[0] * inv_n - mean * mean;
    stats[bh * 2 + 0] = mean;
    stats[bh * 2 + 1] = rsqrtf(var + GN_EPS);
  }
}

__global__ __launch_bounds__(256) void gn_apply_kernel(
    const float* __restrict__ ctx, const float* __restrict__ stats,
    const float* __restrict__ gw, const float* __restrict__ gb,
    float* __restrict__ out) {
  const size_t i = (size_t)blockIdx.x * 256 + threadIdx.x;
  if (i >= (size_t)B * S * D) return;
  const int col = (int)(i % D);
  const int b = (int)(i / ((size_t)S * D));
  const int h = col / DH;
  const float mean = stats[(b * H + h) * 2 + 0];
  const float rstd = stats[(b * H + h) * 2 + 1];
  out[i] = ((ctx[i] - mean) * rstd * gw[col] + gb[col]) * OUT_SCALE;
}

// ---------------------------------------------------------------------------
extern "C" void kernel_launch(void* const* d_in, const int* in_sizes, int n_in,
                              void* d_out, int out_size, void* d_ws, size_t ws_size,
                              hipStream_t stream) {
  const float* x_norm = (const float*)d_in[0];
  const float* lam    = (const float*)d_in[1];
  const float* Cmask  = (const float*)d_in[2];
  const int*   pad    = (const int*)  d_in[3];
  const float* wq_w   = (const float*)d_in[4];
  const float* wq_b   = (const float*)d_in[5];
  const float* wk_w   = (const float*)d_in[6];
  const float* wk_b   = (const float*)d_in[7];
  const float* wv_w   = (const float*)d_in[8];
  const float* wv_b   = (const float*)d_in[9];
  const float* out_w  = (const float*)d_in[10];
  const float* out_b  = (const float*)d_in[11];
  const float* gn_w   = (const float*)d_in[12];
  const float* gn_b   = (const float*)d_in[13];

  float* out   = (float*)d_out;
  float* attn1 = out + (size_t)B * S * D;
  float* attn2 = attn1 + (size_t)B * H * S * S;

  // Workspace layout (floats). ctx/norm overlay q/k projections (dead by then).
  float* qp    = (float*)d_ws;                   // B*S*2D
  float* kp    = qp + (size_t)B * S * 2 * D;     // B*S*2D
  float* vbuf  = kp + (size_t)B * S * 2 * D;     // B*S*D
  float* stats = vbuf + (size_t)B * S * D;       // 2*B*H
  float* ctx   = qp;                             // B*S*D (reuse)
  float* norm  = kp;                             // B*S*D (reuse)

  const int M = B * S;  // 4096

  // Projections
  {
    int waves = (M / 16) * (2 * D / 64);
    gemm_bias_wmma<<<(waves + 3) / 4, 128, 0, stream>>>(x_norm, wq_w, wq_b, qp, M, 2 * D, D);
    gemm_bias_wmma<<<(waves + 3) / 4, 128, 0, stream>>>(x_norm, wk_w, wk_b, kp, M, 2 * D, D);
    int wavesV = (M / 16) * (D / 64);
    gemm_bias_wmma<<<(wavesV + 3) / 4, 128, 0, stream>>>(x_norm, wv_w, wv_b, vbuf, M, D, D);
  }

  // Raw masked scores into d_out attn regions
  {
    int waves = B * H * (S / 16) * (S / 16);
    scores_kernel<<<waves / 4, 128, 0, stream>>>(qp, kp, pad, attn1, attn2);
  }

  // In-place softmax (attn1, attn2 now final outputs)
  softmax_kernel<<<B * H * S, 256, 0, stream>>>(attn1, attn2);

  // Context
  {
    int waves = B * H * (S / 16);
    context_kernel<<<waves / 4, 128, 0, stream>>>(attn1, attn2, Cmask, vbuf, lam, ctx);
  }

  // GroupNorm + scale
  gn_stats_kernel<<<B * H, 256, 0, stream>>>(ctx, stats);
  gn_apply_kernel<<<(int)(((size_t)B * S * D + 255) / 256), 256, 0, stream>>>(
      ctx, stats, gn_w, gn_b, norm);

  // Output projection
  {
    int waves = (M / 16) * (D / 64);
    gemm_bias_wmma<<<(waves + 3) / 4, 128, 0, stream>>>(norm, out_w, out_b, out, M, D, D);
  }
}